// RadiologyLesionAttentionAggregator_214748364981
// MI455X (gfx1250) — compile-verified
//
#include <hip/hip_runtime.h>
#include <math.h>

typedef float v2f __attribute__((ext_vector_type(2)));
typedef float v8f __attribute__((ext_vector_type(8)));

#define DDIM 512
#define NPROJ 768
#define ASTRIDE 516   // 512 + 4 pad -> lane m hits bank (m*4+c)%64, conflict-free
#define PSTRIDE 776   // 768 + 8 pad

// ---- CDNA5 async global->LDS copy (16B per lane), ASYNCcnt tracked ----
__device__ __forceinline__ void async_copy_b128(const float* src, float* lds_dst) {
  unsigned l = (unsigned)(size_t)lds_dst;   // shared-aperture addr truncates to LDS offset
  asm volatile("global_load_async_to_lds_b128 %0, %1, off"
               :: "v"(l), "v"(src) : "memory");
}
__device__ __forceinline__ void async_wait0() {
  asm volatile("s_wait_asynccnt 0x0" ::: "memory");
}

// ---- monotonic float<->uint key for deterministic atomic segment-max ----
__device__ __forceinline__ unsigned fkey(float x) {
  unsigned b = __float_as_uint(x);
  return (b & 0x80000000u) ? ~b : (b | 0x80000000u);
}
__device__ __forceinline__ float funkey(unsigned k) {
  unsigned b = (k & 0x80000000u) ? (k & 0x7FFFFFFFu) : ~k;
  return __uint_as_float(b);
}

// ---- zero-fill scratch (runs every launch; graph-replay safe) ----
__global__ void zero_kernel(float* p, size_t n) {
  size_t i = (size_t)blockIdx.x * blockDim.x + threadIdx.x;
  size_t st = (size_t)gridDim.x * blockDim.x;
  for (; i < n; i += st) p[i] = 0.0f;
}

// ---- fused gather + (feats@W1^T) + tanh + (.@W2^T)+b2 -> scores[E] ----
// block = 128 threads (4 waves), 64 edges. wave = {m-pair, n-half}:
//   mpair = wave>>1 -> M-tiles {2*mpair, 2*mpair+1} (32 edges, dual acc chains)
//   nhalf = wave&1  -> hidden tiles [nhalf*16, nhalf*16+16)
// Each B fragment feeds two WMMAs; per-edge score partials combine in LDS.
__global__ __launch_bounds__(128)
void score_kernel(const float* __restrict__ lesion_x,
                  const long long* __restrict__ les,
                  const float* __restrict__ W1,
                  const float* __restrict__ b1,
                  const float* __restrict__ W2,
                  const float* __restrict__ b2,
                  float* __restrict__ scores, int E) {
  extern __shared__ float smem[];
  float* feats = smem;                 // 64 * ASTRIDE
  float* spart = smem + 64 * ASTRIDE;  // 64 per-edge score partials

  const int tid = threadIdx.x;
  const int e0  = blockIdx.x * 64;

  if (tid < 64) spart[tid] = 0.0f;

  // stage 64 gathered rows via async global->LDS (no VGPR round-trip)
  {
    int row = tid >> 1;
    int cb  = (tid & 1) * 256;
    int e   = e0 + row;
    if (e < E) {
      const float* src = lesion_x + (size_t)les[e] * DDIM + cb;
      float* dst = &feats[row * ASTRIDE + cb];
#pragma unroll
      for (int i = 0; i < 64; ++i) async_copy_b128(src + i * 4, dst + i * 4);
    }
  }
  async_wait0();
  __syncthreads();

  const int wave  = tid >> 5;
  const int lane  = tid & 31;
  const int lm    = lane & 15;
  const int kb    = (lane >> 4) * 2;        // K sub-offset per lane half
  const int mo    = (lane >> 4) * 8;        // M sub-offset per lane half
  const int mpair = wave >> 1;
  const int nhalf = wave & 1;

  const float* arow0 = &feats[(mpair * 32 + lm) * ASTRIDE];
  const float* arow1 = arow0 + 16 * ASTRIDE;

  float s0[8], s1[8];
#pragma unroll
  for (int r = 0; r < 8; ++r) { s0[r] = 0.0f; s1[r] = 0.0f; }

  for (int nt = nhalf * 16; nt < nhalf * 16 + 16; ++nt) {
    const int n0 = nt * 16;
    const float* brow = W1 + (size_t)(n0 + lm) * DDIM;   // B column-major == W1 row
    v8f acc0 = {}, acc1 = {};
#pragma unroll 8
    for (int k0 = 0; k0 < DDIM; k0 += 4) {
      v2f b  = *(const v2f*)(brow + k0 + kb);            // global_load_b64 (L2-resident)
      v2f a0 = *(const v2f*)(arow0 + k0 + kb);           // ds_load_b64
      v2f a1 = *(const v2f*)(arow1 + k0 + kb);
      acc0 = __builtin_amdgcn_wmma_f32_16x16x4_f32(false, a0, false, b,
                                                   (short)0, acc0, false, false);
      acc1 = __builtin_amdgcn_wmma_f32_16x16x4_f32(false, a1, false, b,
                                                   (short)0, acc1, false, false);
    }
    const int n = n0 + lm;
    const float bb = b1[n], w2 = W2[n];
#pragma unroll
    for (int r = 0; r < 8; ++r) {
      s0[r] += tanhf(acc0[r] + bb) * w2;
      s1[r] += tanhf(acc1[r] + bb) * w2;
    }
  }

  // reduce over the 16 lanes of each half (halves = M rows r and r+8)
#pragma unroll
  for (int r = 0; r < 8; ++r) {
    float v0 = s0[r], v1 = s1[r];
    v0 += __shfl_xor(v0, 8, 32); v1 += __shfl_xor(v1, 8, 32);
    v0 += __shfl_xor(v0, 4, 32); v1 += __shfl_xor(v1, 4, 32);
    v0 += __shfl_xor(v0, 2, 32); v1 += __shfl_xor(v1, 2, 32);
    v0 += __shfl_xor(v0, 1, 32); v1 += __shfl_xor(v1, 1, 32);
    s0[r] = v0; s1[r] = v1;
  }
  if (lm == 0) {
#pragma unroll
    for (int r = 0; r < 8; ++r) {
      atomicAdd(&spart[mpair * 32 + mo + r],      s0[r]);   // ds_add_f32
      atomicAdd(&spart[mpair * 32 + 16 + mo + r], s1[r]);
    }
  }
  __syncthreads();
  if (tid < 64) {
    int e = e0 + tid;
    if (e < E) scores[e] = spart[tid] + b2[0];
  }
}

// ---- segment max over patients (deterministic integer atomics) ----
__global__ void segmax_kernel(const float* __restrict__ scores,
                              const long long* __restrict__ pat,
                              unsigned* __restrict__ smaxkey, int E) {
  int e = blockIdx.x * blockDim.x + threadIdx.x;
  if (e < E) atomicMax(&smaxkey[pat[e]], fkey(scores[e]));
}

// ---- ex = exp(score - smax); esum[p] += ex; scores <- ex ----
__global__ void expsum_kernel(float* __restrict__ scores,
                              const long long* __restrict__ pat,
                              const unsigned* __restrict__ smaxkey,
                              float* __restrict__ esum, int E) {
  int e = blockIdx.x * blockDim.x + threadIdx.x;
  if (e < E) {
    long long p = pat[e];
    float ex = expf(scores[e] - funkey(smaxkey[p]));
    scores[e] = ex;
    atomicAdd(&esum[p], ex);
  }
}

// ---- agg[p] += alpha * feats[e]   (block per edge, float4 loads) ----
__global__ __launch_bounds__(128)
void agg_kernel(const float* __restrict__ lesion_x,
                const long long* __restrict__ pat,
                const long long* __restrict__ les,
                const float* __restrict__ ex,
                const float* __restrict__ esum,
                float* __restrict__ agg, int E) {
  int e = blockIdx.x;
  if (e >= E) return;
  long long p = pat[e];
  float alpha = ex[e] / fmaxf(esum[p], 1e-12f);
  const float4* src = (const float4*)(lesion_x + (size_t)les[e] * DDIM);
  float* dst = agg + (size_t)p * DDIM;
  int t = threadIdx.x;                 // 128 threads * 4 cols = 512
  float4 v = src[t];
  atomicAdd(dst + t * 4 + 0, v.x * alpha);
  atomicAdd(dst + t * 4 + 1, v.y * alpha);
  atomicAdd(dst + t * 4 + 2, v.z * alpha);
  atomicAdd(dst + t * 4 + 3, v.w * alpha);
}

// ---- proj = agg @ Wp^T + bp, then fused LayerNorm over 768 ----
// block = 128 threads; block owns 16 patients; 4 waves split the 48 N-tiles.
__global__ __launch_bounds__(128)
void proj_ln_kernel(const float* __restrict__ agg,
                    const float* __restrict__ Wp,
                    const float* __restrict__ bp,
                    const float* __restrict__ gamma,
                    const float* __restrict__ beta,
                    float* __restrict__ out, int P) {
  extern __shared__ float smem[];
  float* atile = smem;                        // 16*ASTRIDE
  float* ptile = atile + 16 * ASTRIDE;        // 16*PSTRIDE
  float* psum  = ptile + 16 * PSTRIDE;        // 16*8
  float* psq   = psum + 128;                  // 16*8
  float* mu_s  = psq + 128;                   // 16
  float* rs_s  = mu_s + 16;                   // 16

  const int tid = threadIdx.x;
  const int p0  = blockIdx.x * 16;

  // stage 16 agg rows via async global->LDS: 8 threads/row, 16B each
  {
    int row = tid >> 3, seg = (tid & 7) * 64;
    if (p0 + row < P) {
      const float* src = agg + (size_t)(p0 + row) * DDIM + seg;
      float* dst = &atile[row * ASTRIDE + seg];
#pragma unroll
      for (int i = 0; i < 16; ++i) async_copy_b128(src + i * 4, dst + i * 4);
    }
  }
  async_wait0();
  __syncthreads();

  const int wave = tid >> 5, lane = tid & 31;
  const int lm = lane & 15, kb = (lane >> 4) * 2, mo = (lane >> 4) * 8;
  const float* arow = &atile[lm * ASTRIDE];

  for (int nt = wave; nt < NPROJ / 16; nt += 4) {
    const int n0 = nt * 16;
    const float* brow = Wp + (size_t)(n0 + lm) * DDIM;
    v8f acc = {};
#pragma unroll 8
    for (int k0 = 0; k0 < DDIM; k0 += 4) {
      v2f a = *(const v2f*)(arow + k0 + kb);
      v2f b = *(const v2f*)(brow + k0 + kb);
      acc = __builtin_amdgcn_wmma_f32_16x16x4_f32(false, a, false, b,
                                                  (short)0, acc, false, false);
    }
    const int n = n0 + lm;
    const float bias = bp[n];
#pragma unroll
    for (int r = 0; r < 8; ++r) ptile[(mo + r) * PSTRIDE + n] = acc[r] + bias;
  }
  __syncthreads();

  // LayerNorm over 768: 8 partial sums per row
  {
    int row = tid >> 3, seg = tid & 7;
    const float* pr = &ptile[row * PSTRIDE + seg * 96];
    float s = 0.0f, sq = 0.0f;
    for (int i = 0; i < 96; ++i) { float v = pr[i]; s += v; sq += v * v; }
    psum[row * 8 + seg] = s;
    psq[row * 8 + seg]  = sq;
  }
  __syncthreads();
  if (tid < 16) {
    float s = 0.0f, sq = 0.0f;
    for (int i = 0; i < 8; ++i) { s += psum[tid * 8 + i]; sq += psq[tid * 8 + i]; }
    float mu  = s * (1.0f / NPROJ);
    float var = sq * (1.0f / NPROJ) - mu * mu;
    mu_s[tid] = mu;
    rs_s[tid] = rsqrtf(var + 1e-5f);
  }
  __syncthreads();
  for (int idx = tid; idx < 16 * NPROJ; idx += 128) {
    int row = idx / NPROJ, c = idx - row * NPROJ;
    if (p0 + row < P) {
      float v = (ptile[row * PSTRIDE + c] - mu_s[row]) * rs_s[row] * gamma[c] + beta[c];
      out[(size_t)(p0 + row) * NPROJ + c] = v;
    }
  }
}

extern "C" void kernel_launch(void* const* d_in, const int* in_sizes, int n_in,
                              void* d_out, int out_size, void* d_ws, size_t ws_size,
                              hipStream_t stream) {
  const float*     lesion_x = (const float*)d_in[0];
  const long long* edge     = (const long long*)d_in[1];   // int64 per reference
  const float*     W1    = (const float*)d_in[3];
  const float*     b1    = (const float*)d_in[4];
  const float*     W2    = (const float*)d_in[5];
  const float*     b2    = (const float*)d_in[6];
  const float*     Wp    = (const float*)d_in[7];
  const float*     bp    = (const float*)d_in[8];
  const float*     gamma = (const float*)d_in[9];
  const float*     beta  = (const float*)d_in[10];
  float*           out   = (float*)d_out;

  const int E = in_sizes[1] / 2;
  const int P = out_size / NPROJ;
  const long long* pat = edge;
  const long long* les = edge + E;

  // workspace layout (floats): scores[E] | smaxkey[P] | esum[P] | agg[P*512]
  float*    scores  = (float*)d_ws;
  unsigned* smaxkey = (unsigned*)(scores + E);
  float*    esum    = (float*)(smaxkey + P);
  float*    agg     = esum + P;

  // 1) zero smaxkey/esum/agg (key 0 == "-inf")
  size_t nz = (size_t)P * 2 + (size_t)P * DDIM;
  zero_kernel<<<1024, 256, 0, stream>>>((float*)smaxkey, nz);

  // 2) fused gather + GEMM + tanh + score
  size_t smemA = (size_t)(64 * ASTRIDE + 64) * sizeof(float);
  score_kernel<<<(E + 63) / 64, 128, smemA, stream>>>(
      lesion_x, les, W1, b1, W2, b2, scores, E);

  // 3) segment max
  segmax_kernel<<<(E + 255) / 256, 256, 0, stream>>>(scores, pat, smaxkey, E);

  // 4) exp + segment sum
  expsum_kernel<<<(E + 255) / 256, 256, 0, stream>>>(scores, pat, smaxkey, esum, E);

  // 5) alpha-weighted aggregation
  agg_kernel<<<E, 128, 0, stream>>>(lesion_x, pat, les, scores, esum, agg, E);

  // 6) projection GEMM + fused LayerNorm
  size_t smemF = (size_t)(16 * ASTRIDE + 16 * PSTRIDE + 128 + 128 + 32) * sizeof(float);
  proj_ln_kernel<<<(P + 15) / 16, 128, smemF, stream>>>(agg, Wp, bp, gamma, beta, out, P);
}